// LEnsemble_67345087201711
// MI455X (gfx1250) — compile-verified
//
#include <hip/hip_runtime.h>
#include <hip/hip_bf16.h>
#include <math.h>

// ---------------------------------------------------------------------------
// LEnsemble logdet for MI455X (gfx1250, wave32, WMMA).
//   L = B^T B + eps*I                         (WMMA f32 16x16x4 GEMM)
//   Lb = mask(L, x[b]) (+1 diag inactive)     (elementwise build)
//   logdet via BLOCKED in-place Cholesky      (1 workgroup / matrix):
//       POTRF(32x32 in LDS, staged via ASYNC global->LDS)
//       -> TRSM(panel, regs) -> SYRK trailing via WMMA
//   out[b] = 2*sumlog(diag chol Lb) - 2*sumlog(diag chol (L+I))
// ---------------------------------------------------------------------------

typedef __attribute__((ext_vector_type(2))) float v2f;
typedef __attribute__((ext_vector_type(8))) float v8f;

#define LN   512
#define SLAB (512 * 512)          // floats per matrix
#define NMAT 129                  // 128 batches + 1 normalizer (L+I)
#define NB   32                   // Cholesky block size
#define EPSV 1e-8f

// --------------------------------------------------------------------------
// Kernel 1: L = B^T B + eps*I, f32 WMMA. One wave per 16x16 tile.
// A-frag (16x4): lane l (h=l/16, m=l%16) holds A[m, 2h+v], v=0,1
// B-frag (4x16): lane l holds             B[2h+v, m]
// C/D   (16x16): vgpr r, lane l holds     D[r + 8h, m]
// --------------------------------------------------------------------------
__global__ void __launch_bounds__(128)
le_gemm_btb(const float* __restrict__ Bsrc, float* __restrict__ L)
{
    const int wave = threadIdx.x >> 5;
    const int lane = threadIdx.x & 31;
    const int tile = blockIdx.x * 4 + wave;      // 0..1023
    const int ti = tile >> 5;
    const int tj = tile & 31;
    const int h  = lane >> 4;
    const int m  = lane & 15;

    const float* colA = Bsrc + ti * 16 + m;
    const float* colB = Bsrc + tj * 16 + m;

    v8f c = {};
    for (int k0 = 0; k0 < LN; k0 += 4) {
        const int r0 = (k0 + 2 * h) * LN;
        __builtin_prefetch(colA + r0 + 32 * LN, 0, 1);
        v2f a, b;
        a.x = colA[r0];
        a.y = colA[r0 + LN];
        b.x = colB[r0];
        b.y = colB[r0 + LN];
        c = __builtin_amdgcn_wmma_f32_16x16x4_f32(
                false, a, false, b, (short)0, c, false, false);
    }

    const int col = tj * 16 + m;
    #pragma unroll
    for (int r = 0; r < 8; ++r) {
        const int row = ti * 16 + r + 8 * h;
        float v = c[r];
        if (row == col) v += EPSV;
        L[row * LN + col] = v;
    }
}

// --------------------------------------------------------------------------
// Kernel 2: build masked matrices (and L+I as id 128) into slabs.
// --------------------------------------------------------------------------
__global__ void __launch_bounds__(256)
le_build_mats(const float* __restrict__ L, const int* __restrict__ x,
              float* __restrict__ slabs, int mat_base)
{
    const int idx = blockIdx.x * 256 + threadIdx.x;   // 0..262143
    const int r  = idx >> 9;
    const int cc = idx & (LN - 1);
    const int id = mat_base + blockIdx.y;
    float* A = slabs + (size_t)blockIdx.y * SLAB;

    const float lv = L[idx];
    float v;
    if (id < 128) {
        const float mi = (x[id * LN + r]  != 0) ? 1.0f : 0.0f;
        const float mj = (x[id * LN + cc] != 0) ? 1.0f : 0.0f;
        v = lv * mi * mj;
        if (r == cc && mj == 0.0f) v = 1.0f;
    } else {
        v = lv + ((r == cc) ? 1.0f : 0.0f);
    }
    A[idx] = v;
}

// --------------------------------------------------------------------------
// Kernel 3: blocked in-place Cholesky, one 256-thread workgroup per matrix.
//   phase 1: POTRF on 32x32 diag block in LDS (padded stride 33),
//            block staged from global via GLOBAL_LOAD_ASYNC_TO_LDS_B32
//   phase 2: TRSM  P * Lkk^T = A_panel, row-parallel, row in registers
//   phase 3: SYRK  trailing -= P P^T, 16x16 tiles on WMMA, one tile/wave
// Only the lower triangle (at 16-tile granularity) is maintained; upper
// garbage is never read.
// --------------------------------------------------------------------------
__global__ void __launch_bounds__(256)
le_chol_blocked(float* __restrict__ slabs)
{
    float* A = slabs + (size_t)blockIdx.x * SLAB;
    __shared__ float Lkk[NB * (NB + 1)];   // 32x32, stride 33
    __shared__ float sinv;
    const int tid  = threadIdx.x;
    const int wave = tid >> 5;
    const int lane = tid & 31;
    const int h    = lane >> 4;
    const int m    = lane & 15;

    for (int k = 0; k < LN; k += NB) {
        // ---- async-stage diagonal block into LDS (ASYNCcnt path) ----
        // generic pointer to __shared__ carries the LDS byte offset in its
        // low 32 bits (LDS aperture truncation), which is what the async
        // instruction's LDS-address VGPR expects.
        for (int idx = tid; idx < NB * NB; idx += 256) {
            const int r = idx >> 5, c = idx & (NB - 1);
            unsigned lds_off = (unsigned)(size_t)(&Lkk[r * (NB + 1) + c]);
            const float* gp = &A[(k + r) * LN + (k + c)];
            asm volatile("global_load_async_to_lds_b32 %0, %1, off"
                         :: "v"(lds_off), "v"(gp)
                         : "memory");
        }
        asm volatile("s_wait_asynccnt 0" ::: "memory");
        __syncthreads();

        // ---- POTRF (unblocked, in LDS) ----
        for (int j = 0; j < NB; ++j) {
            if (tid == 0) {
                const float d = sqrtf(Lkk[j * (NB + 1) + j]);
                Lkk[j * (NB + 1) + j] = d;
                sinv = 1.0f / d;
            }
            __syncthreads();
            if (tid > j && tid < NB)
                Lkk[tid * (NB + 1) + j] *= sinv;
            __syncthreads();
            if (tid > j && tid < NB) {
                const float lij = Lkk[tid * (NB + 1) + j];
                for (int c = j + 1; c <= tid; ++c)
                    Lkk[tid * (NB + 1) + c] -= lij * Lkk[c * (NB + 1) + j];
            }
            __syncthreads();
        }

        // ---- write diag block back (diag needed by logdet kernel) ----
        for (int idx = tid; idx < NB * NB; idx += 256) {
            const int r = idx >> 5, c = idx & (NB - 1);
            A[(k + r) * LN + (k + c)] = Lkk[r * (NB + 1) + c];
        }
        __syncthreads();

        // ---- TRSM: P[i,:] solves P * Lkk^T = A_panel, rows k+NB..511 ----
        for (int i = k + NB + tid; i < LN; i += 256) {
            float p[NB];
            float* __restrict__ arow = A + i * LN + k;
            for (int j = 0; j < NB; ++j) {
                float v = arow[j];
                for (int t = 0; t < j; ++t)
                    v -= p[t] * Lkk[j * (NB + 1) + t];
                p[j] = v / Lkk[j * (NB + 1) + j];
            }
            for (int j = 0; j < NB; ++j) arow[j] = p[j];
        }
        __threadfence();
        __syncthreads();

        // ---- SYRK: trailing(i,j) -= P_i P_j^T via WMMA, lower 16-tiles ----
        const int t0 = k + NB;
        const int nt = (LN - t0) >> 4;           // 16-tiles per trailing dim
        if (nt > 0) {
            const int ntiles = nt * (nt + 1) / 2;
            for (int t = wave; t < ntiles; t += 8) {
                // t -> (a,b), a >= b, t = a(a+1)/2 + b
                int a = (int)((sqrtf(8.0f * (float)t + 1.0f) - 1.0f) * 0.5f);
                while ((a + 1) * (a + 2) / 2 <= t) ++a;
                while (a * (a + 1) / 2 > t) --a;
                const int b  = t - a * (a + 1) / 2;
                const int r0 = t0 + a * 16;
                const int c0 = t0 + b * 16;

                v8f cf;
                #pragma unroll
                for (int r = 0; r < 8; ++r)
                    cf[r] = A[(r0 + r + 8 * h) * LN + c0 + m];

                const float* Pa = A + (r0 + m) * LN + k;  // row r0+m, cols k..k+31
                const float* Pb = A + (c0 + m) * LN + k;  // row c0+m, cols k..k+31
                #pragma unroll
                for (int kk = 0; kk < NB; kk += 4) {
                    v2f af, bf;
                    // negate A-frag: cf + (-P_a)*P_b^T  == cf - P_a P_b^T
                    // (f32 WMMA forbids NEG[0]/NEG[1], so fold sign here)
                    af.x = -Pa[kk + 2 * h];
                    af.y = -Pa[kk + 2 * h + 1];
                    bf.x =  Pb[kk + 2 * h];
                    bf.y =  Pb[kk + 2 * h + 1];
                    cf = __builtin_amdgcn_wmma_f32_16x16x4_f32(
                            false, af, false, bf, (short)0, cf, false, false);
                }

                #pragma unroll
                for (int r = 0; r < 8; ++r)
                    A[(r0 + r + 8 * h) * LN + c0 + m] = cf[r];
            }
        }
        __threadfence();
        __syncthreads();
    }
}

// --------------------------------------------------------------------------
// Kernel 4: logdet = 2 * sum(log(diag)), block reduction per matrix.
// --------------------------------------------------------------------------
__global__ void __launch_bounds__(256)
le_logdet_sum(const float* __restrict__ slabs, float* __restrict__ logdets,
              int mat_base)
{
    const float* A = slabs + (size_t)blockIdx.x * SLAB;
    __shared__ float red[256];
    float s = 0.0f;
    for (int i = threadIdx.x; i < LN; i += 256)
        s += logf(A[i * LN + i]);
    red[threadIdx.x] = s;
    __syncthreads();
    for (int off = 128; off > 0; off >>= 1) {
        if (threadIdx.x < off) red[threadIdx.x] += red[threadIdx.x + off];
        __syncthreads();
    }
    if (threadIdx.x == 0)
        logdets[mat_base + blockIdx.x] = 2.0f * red[0];
}

__global__ void le_finalize(const float* __restrict__ logdets,
                            float* __restrict__ out)
{
    const int b = threadIdx.x;      // 128 threads
    out[b] = logdets[b] - logdets[128];
}

// --------------------------------------------------------------------------
// Host launcher. ws layout (floats):
//   [0, SLAB)        : L
//   [SLAB, SLAB+256) : logdets (129 used)
//   [SLAB+256, ...)  : nslab matrix slabs (chunked if ws is small)
// --------------------------------------------------------------------------
extern "C" void kernel_launch(void* const* d_in, const int* in_sizes, int n_in,
                              void* d_out, int out_size, void* d_ws, size_t ws_size,
                              hipStream_t stream)
{
    (void)in_sizes; (void)n_in; (void)out_size;
    const int*   x    = (const int*)d_in[0];     // (128, 512) int32
    const float* Bmat = (const float*)d_in[1];   // (512, 512) f32
    float* out = (float*)d_out;                  // (128,) f32
    float* ws  = (float*)d_ws;

    float* L       = ws;
    float* logdets = ws + SLAB;
    float* slabs   = ws + SLAB + 256;

    long avail = (long)(ws_size / sizeof(float)) - (long)SLAB - 256;
    int nslab = (int)(avail / (long)SLAB);
    if (nslab < 1)    nslab = 1;
    if (nslab > NMAT) nslab = NMAT;

    le_gemm_btb<<<256, 128, 0, stream>>>(Bmat, L);

    for (int m0 = 0; m0 < NMAT; ) {
        int cnt = NMAT - m0;
        if (cnt > nslab) cnt = nslab;
        le_build_mats<<<dim3(1024, cnt), 256, 0, stream>>>(L, x, slabs, m0);
        le_chol_blocked<<<cnt, 256, 0, stream>>>(slabs);
        le_logdet_sum<<<cnt, 256, 0, stream>>>(slabs, logdets, m0);
        m0 += cnt;
    }

    le_finalize<<<1, 128, 0, stream>>>(logdets, out);
}